// TransformerDecoder_55164559950058
// MI455X (gfx1250) — compile-verified
//
#include <hip/hip_runtime.h>
#include <hip/hip_bf16.h>

// Problem constants (from the reference)
#define BB    16
#define QQ    300
#define SS    16
#define HID   256
#define W0    100
#define H0    100
#define TTOK  13125          // 100*100 + 50*50 + 25*25
#define NROW  (BB * 3 * SS)  // 768 distinct feature rows actually consumed
#define SSCALE 0.077f

typedef float v2f __attribute__((ext_vector_type(2)));
typedef float v8f __attribute__((ext_vector_type(8)));

// CDNA5 has a hardware V_TANH_F32 transcendental; use it when the builtin
// exists (clang-22+ for gfx1250), else fall back to libm tanhf.
__device__ __forceinline__ float hw_tanh(float x) {
#if __has_builtin(__builtin_amdgcn_tanhf)
  return __builtin_amdgcn_tanhf(x);
#else
  return tanhf(x);
#endif
}

// ---------------------------------------------------------------------------
// Kernel 1: bilinear sampling of the 768 (b, lvl, s) rows at LEVEL 0 only.
// One block per row, one channel per thread (coalesced over C).
// ---------------------------------------------------------------------------
__global__ __launch_bounds__(256) void sample_kernel(
    const float* __restrict__ rp,      // [B,Q,8]
    const float* __restrict__ memory,  // [B,TTOK,HID]
    float* __restrict__ g)             // [NROW,HID]
{
  const int rowid = blockIdx.x;            // (b*3 + lvl)*16 + s
  const int s   = rowid & 15;
  const int bl  = rowid >> 4;
  const int lvl = bl % 3;                  // doubles as the QUERY index (gather quirk)
  const int b   = bl / 3;

  const float t = (float)s * (1.0f / 15.0f);
  const float* c = rp + (size_t)(b * QQ + lvl) * 8;
  const float px = ((c[0] * t + c[1]) * t + c[2]) * t + c[3];
  const float py = ((c[4] * t + c[5]) * t + c[6]) * t + c[7];
  const float gx = 2.0f * px - 1.0f;       // 2*(poly - 0.5)
  const float gy = 2.0f * py - 1.0f;

  // grid_sample, align_corners=False, padding zeros; level 0 is 100x100
  const float x = (gx + 1.0f) * (W0 * 0.5f) - 0.5f;
  const float y = (gy + 1.0f) * (H0 * 0.5f) - 0.5f;
  const float x0f = floorf(x), y0f = floorf(y);
  const int x0 = (int)x0f, y0 = (int)y0f;
  const int x1 = x0 + 1,   y1 = y0 + 1;
  const float wx1 = x - x0f, wx0 = 1.0f - wx1;
  const float wy1 = y - y0f, wy0 = 1.0f - wy1;

  const int ch = threadIdx.x;
  const float* mb = memory + (size_t)b * TTOK * HID;  // level-0 tokens start at 0

  auto fetch = [&](int xi, int yi) -> float {
    if (xi < 0 || xi >= W0 || yi < 0 || yi >= H0) return 0.0f;
    return mb[(size_t)(yi * W0 + xi) * HID + ch];     // feat[b,ch,yi,xi] = memory[b, yi*W+xi, ch]
  };

  const float v = fetch(x0, y0) * (wx0 * wy0)
                + fetch(x1, y0) * (wx1 * wy0)
                + fetch(x0, y1) * (wx0 * wy1)
                + fetch(x1, y1) * (wx1 * wy1);

  g[(size_t)rowid * HID + ch] = v;
}

// ---------------------------------------------------------------------------
// Kernel 2: fused 2-layer MLP on the 768 rows, f32 WMMA (V_WMMA_F32_16X16X4).
// 48 blocks x 256 threads (8 waves); block = one 16-row M tile.
//   Layer 1: out16x256 = tanh(g_tile @ w1 + b1)  -> LDS
//   Layer 2: toff16x4  = SSCALE * tanh(h1 @ w2 + b2)
// f32 WMMA register layouts (ISA 7.12.2):
//   A 16x4: lane L<16 -> M=L, VGPR{0,1}=K{0,1}; lane>=16 -> M=L-16, K{2,3}
//   B 4x16: lane L<16 -> N=L, VGPR{0,1}=K{0,1}; lane>=16 -> N=L-16, K{2,3}
//   C/D 16x16: VGPR r: lanes<16 -> (M=r, N=lane); lanes>=16 -> (M=r+8, N=lane-16)
// ---------------------------------------------------------------------------
#define LDS_STRIDE 258

__global__ __launch_bounds__(256) void mlp_kernel(
    const float* __restrict__ g,    // [NROW,HID]
    const float* __restrict__ w1,   // [HID,HID]
    const float* __restrict__ b1,   // [HID]
    const float* __restrict__ w2,   // [HID,4]
    const float* __restrict__ b2,   // [4]
    float* __restrict__ toff)       // [NROW,4]
{
  __shared__ float h1s[16 * LDS_STRIDE];

  const int mtile = blockIdx.x;        // 0..47
  const int lane  = threadIdx.x & 31;
  const int wv    = threadIdx.x >> 5;  // 0..7
  const int row   = lane & 15;         // M (for A) / N (for B,C,D) selector
  const int hi    = lane >> 4;         // K-half selector

  const float* grow = g + (size_t)(mtile * 16 + row) * HID;

  // ---- Layer 1: each wave computes two 16-col N tiles ----
  for (int nt = 0; nt < 2; ++nt) {
    const int ncol = (wv + nt * 8) * 16 + row;     // output column
    v8f acc;
    {
      const float bb = b1[ncol];                   // same N for all 8 acc regs of a lane
      #pragma unroll
      for (int i = 0; i < 8; ++i) acc[i] = bb;
    }
    #pragma unroll 4
    for (int k0 = 0; k0 < HID; k0 += 4) {
      const int k = k0 + 2 * hi;
      v2f a = *(const v2f*)(grow + k);             // A[M=row, K=k..k+1]
      v2f b;
      b[0] = w1[(size_t)k * HID + ncol];           // B[K=k,   N=ncol]
      b[1] = w1[(size_t)(k + 1) * HID + ncol];     // B[K=k+1, N=ncol]
      acc = __builtin_amdgcn_wmma_f32_16x16x4_f32(
          false, a, false, b, (short)0, acc, false, false);
    }
    #pragma unroll
    for (int i = 0; i < 8; ++i)
      h1s[(i + 8 * hi) * LDS_STRIDE + ncol] = hw_tanh(acc[i]);
  }

  __syncthreads();

  // ---- Layer 2: wave 0 only (EXEC all-ones within the wave) ----
  if (wv == 0) {
    const int  ncol = row;              // 0..15; only 0..3 are real columns
    const bool live = (ncol < 4);
    v8f acc;
    {
      const float bb = live ? b2[ncol] : 0.0f;
      #pragma unroll
      for (int i = 0; i < 8; ++i) acc[i] = bb;
    }
    #pragma unroll 4
    for (int k0 = 0; k0 < HID; k0 += 4) {
      const int k = k0 + 2 * hi;
      v2f a = *(const v2f*)(&h1s[row * LDS_STRIDE + k]);   // A[M=row, K=k..k+1]
      v2f b;
      b[0] = live ? w2[(size_t)k * 4 + ncol] : 0.0f;       // zero-padded w2 column
      b[1] = live ? w2[(size_t)(k + 1) * 4 + ncol] : 0.0f;
      acc = __builtin_amdgcn_wmma_f32_16x16x4_f32(
          false, a, false, b, (short)0, acc, false, false);
    }
    if (live) {
      #pragma unroll
      for (int i = 0; i < 8; ++i) {
        const int m = i + 8 * hi;
        toff[(size_t)(mtile * 16 + m) * 4 + ncol] = SSCALE * hw_tanh(acc[i]);
      }
    }
  }
}

// ---------------------------------------------------------------------------
// Kernel 3: final assembly. out[b,q,s,np,xy] = toff[b,ref_levels[b,q],s,np,xy]
//                                            + sp[b,q,s,xy]
// 76800 threads, one (b,q,s) each, float4 store (writes every output element).
// ---------------------------------------------------------------------------
__global__ __launch_bounds__(256) void assemble_kernel(
    const float* __restrict__ rp,     // [B,Q,8]
    const int*   __restrict__ lvls,   // [B,Q]
    const float* __restrict__ toff,   // [NROW,4]
    float* __restrict__ out)          // [B,Q,S*2,2]
{
  const int i  = blockIdx.x * 256 + threadIdx.x;   // 0..76799 == b*Q*S + q*S + s
  const int s  = i & 15;
  const int bq = i >> 4;                           // b*Q + q
  const int b  = bq / QQ;

  const float t = (float)s * (1.0f / 15.0f);
  const float* c = rp + (size_t)bq * 8;
  const float spx = 2.0f * (((c[0] * t + c[1]) * t + c[2]) * t + c[3]) - 1.0f;
  const float spy = 2.0f * (((c[4] * t + c[5]) * t + c[6]) * t + c[7]) - 1.0f;

  int lvl = lvls[bq];
  lvl = lvl < 0 ? 0 : (lvl > 2 ? 2 : lvl);

  const float4 tv = *(const float4*)(toff + ((size_t)(b * 3 + lvl) * 16 + s) * 4);
  float4 o;
  o.x = tv.x + spx;   // np=0, x
  o.y = tv.y + spy;   // np=0, y
  o.z = tv.z + spx;   // np=1, x
  o.w = tv.w + spy;   // np=1, y
  *(float4*)(out + (size_t)i * 4) = o;
}

// ---------------------------------------------------------------------------
extern "C" void kernel_launch(void* const* d_in, const int* in_sizes, int n_in,
                              void* d_out, int out_size, void* d_ws, size_t ws_size,
                              hipStream_t stream) {
  const float* rp     = (const float*)d_in[0];  // ref_polys [16,300,8]
  const float* memory = (const float*)d_in[1];  // [16,13125,256]
  const float* w1     = (const float*)d_in[2];  // [256,256]
  const float* b1     = (const float*)d_in[3];  // [256]
  const float* w2     = (const float*)d_in[4];  // [256,4]
  const float* b2     = (const float*)d_in[5];  // [4]
  const int*   lvls   = (const int*)d_in[6];    // [16,300]
  float* out = (float*)d_out;

  float* g    = (float*)d_ws;                                  // [768,256]
  float* toff = (float*)((char*)d_ws + (size_t)NROW * HID * sizeof(float)); // [768,4]

  sample_kernel  <<<NROW,           256, 0, stream>>>(rp, memory, g);
  mlp_kernel     <<<NROW / 16,      256, 0, stream>>>(g, w1, b1, w2, b2, toff);
  assemble_kernel<<<(BB*QQ*SS)/256, 256, 0, stream>>>(rp, lvls, toff, out);
}